// Net_24584392802821
// MI455X (gfx1250) — compile-verified
//
#include <hip/hip_runtime.h>
#include <hip/hip_bf16.h>
#include <stddef.h>

// ---------------------------------------------------------------------------
// ChebConv (K=6) on MI455X / gfx1250 (wave32, WMMA).
//   T0 = x ; T1 = L x ; Tk = 2 L T_{k-1} - T_{k-2}
//   L z = sum_{e: dst=i} (-w_e) z[src_e]  -  z[i]        (norm = -w, loop = -1)
//   out = sum_k T_k @ W_k + bias
// Dense GEMMs use V_WMMA_F32_16X16X4_F32 (exact f32). Sparse prop uses
// relaxed agent-scope f32 atomics (global_atomic_add_f32) that resolve in
// the 192MB L2 (entire working set is L2-resident).
// W_k is staged in LDS in K-pair-interleaved form so each B fragment is one
// aligned ds_load_b64 (no register-pair shuffling in the WMMA loop).
// ---------------------------------------------------------------------------

typedef __attribute__((ext_vector_type(2))) float v2f;
typedef __attribute__((ext_vector_type(8))) float v8f;

#define FDIM 64

// z_out = alpha * a + beta * b   (vectorized float4 over N*64 elements)
__global__ __launch_bounds__(256) void cheb_init_combine(
    float* __restrict__ zout, const float* __restrict__ a, float alpha,
    const float* __restrict__ b, float beta, int nquads)
{
    int i = blockIdx.x * blockDim.x + threadIdx.x;
    if (i >= nquads) return;
    const float4 va = ((const float4*)a)[i];
    const float4 vb = ((const float4*)b)[i];
    float4 r;
    r.x = alpha * va.x + beta * vb.x;
    r.y = alpha * va.y + beta * vb.y;
    r.z = alpha * va.z + beta * vb.z;
    r.w = alpha * va.w + beta * vb.w;
    ((float4*)zout)[i] = r;
}

// zout[dst] += (coef * w_e) * zin[src]  ; 16 lanes per edge, float4 each.
__global__ __launch_bounds__(256) void cheb_edge_scatter(
    float* __restrict__ zout, const float* __restrict__ zin,
    const int* __restrict__ src, const int* __restrict__ dst,
    const float* __restrict__ ew, float coef, int nedges)
{
    int tid  = blockIdx.x * blockDim.x + threadIdx.x;
    int e    = tid >> 4;
    int part = tid & 15;
    if (e >= nedges) return;
    int   s = src[e];
    int   d = dst[e];
    float c = coef * ew[e];
    const float4 v = *(const float4*)(zin + (size_t)s * FDIM + part * 4);
    float* o = zout + (size_t)d * FDIM + part * 4;
    __hip_atomic_fetch_add(o + 0, c * v.x, __ATOMIC_RELAXED, __HIP_MEMORY_SCOPE_AGENT);
    __hip_atomic_fetch_add(o + 1, c * v.y, __ATOMIC_RELAXED, __HIP_MEMORY_SCOPE_AGENT);
    __hip_atomic_fetch_add(o + 2, c * v.z, __ATOMIC_RELAXED, __HIP_MEMORY_SCOPE_AGENT);
    __hip_atomic_fetch_add(o + 3, c * v.w, __ATOMIC_RELAXED, __HIP_MEMORY_SCOPE_AGENT);
}

// out[16x64 strip] (+)= T[16x64 strip] @ Wk[64x64] using f32 WMMA 16x16x4.
// 8 waves per block, one 16-row strip per wave; Wk staged in LDS.
//
// Fragment layouts (CDNA5 ISA 7.12.2, wave32):
//   A 16x4 f32 : lane(0..15)=row M, v0/v1 = K=(2*hi)+0/+1 where hi=lane>>4
//   B 4x16 f32 : lane&15 = col N,   v0/v1 = K=(2*hi)+0/+1
//   C 16x16 f32: vgpr j, lanes0-15 -> (M=j, N=lane), lanes16-31 -> (M=j+8)
//
// LDS W layout (pair-interleaved): lW[(k>>1)*128 + col*2 + (k&1)]
//   -> B fragment (K=2p, 2p+1 at column c) is one 8-byte ds_load_b64.
__global__ __launch_bounds__(256) void cheb_gemm_wmma(
    const float* __restrict__ T, const float* __restrict__ Wk,
    const float* __restrict__ bias, float* __restrict__ out,
    int initMode, int nrows)
{
    __shared__ float lW[FDIM * FDIM];   // 16 KB, pair-interleaved
    for (int i = threadIdx.x; i < FDIM * FDIM; i += 256) {
        const int k = i >> 6;           // source row (K index)
        const int c = i & 63;           // source col (N index)
        lW[(k >> 1) * 128 + c * 2 + (k & 1)] = Wk[i];
    }
    __syncthreads();

    const int wave  = threadIdx.x >> 5;
    const int lane  = threadIdx.x & 31;
    const int strip = blockIdx.x * 8 + wave;
    const int mbase = strip * 16;
    if (mbase >= nrows) return;       // wave-uniform: EXEC stays all-1s

    const int nl = lane & 15;
    const int hi = lane >> 4;
    const int kb = hi * 2;
    const int arow = mbase + nl;      // A-fragment row for this lane

    v8f c[4];
    if (initMode) {
#pragma unroll
        for (int t = 0; t < 4; ++t) {
            const float bv = bias[t * 16 + nl];
#pragma unroll
            for (int j = 0; j < 8; ++j) c[t][j] = bv;
        }
    } else {
#pragma unroll
        for (int t = 0; t < 4; ++t)
#pragma unroll
            for (int j = 0; j < 8; ++j)
                c[t][j] = out[(size_t)(mbase + j + hi * 8) * FDIM + t * 16 + nl];
    }

#pragma unroll
    for (int kk = 0; kk < 16; ++kk) {     // K = 64 in steps of 4
        const int k0 = kk * 4;
        const v2f a = *(const v2f*)(T + (size_t)arow * FDIM + k0 + kb);
        // pair index for this half-wave: p = (k0+kb)/2 = kk*2 + hi
        const int pbase = (kk * 2 + hi) * 128;
#pragma unroll
        for (int t = 0; t < 4; ++t) {
            const v2f b = *(const v2f*)(lW + pbase + (t * 16 + nl) * 2);
            c[t] = __builtin_amdgcn_wmma_f32_16x16x4_f32(
                false, a, false, b, (short)0, c[t], false, false);
        }
    }

#pragma unroll
    for (int t = 0; t < 4; ++t)
#pragma unroll
        for (int j = 0; j < 8; ++j)
            out[(size_t)(mbase + j + hi * 8) * FDIM + t * 16 + nl] = c[t][j];
}

extern "C" void kernel_launch(void* const* d_in, const int* in_sizes, int n_in,
                              void* d_out, int out_size, void* d_ws, size_t ws_size,
                              hipStream_t stream)
{
    const float* x    = (const float*)d_in[0];       // [N,64]
    const int*   idx  = (const int*)  d_in[1];       // [2,E]
    const float* ew   = (const float*)d_in[2];       // [E]
    const float* W    = (const float*)d_in[3];       // [6,64,64]
    const float* bias = (const float*)d_in[4];       // [64]
    float*       out  = (float*)d_out;               // [N,64]

    const int N = in_sizes[0] / FDIM;
    const int E = in_sizes[2];
    const int* src = idx;
    const int* dst = idx + E;

    const size_t NF = (size_t)N * FDIM;
    float* b0 = (float*)d_ws;
    float* b1 = b0 + NF;
    float* b2 = b1 + NF;
    float* bufs[3] = {b0, b1, b2};

    const int nquads     = (int)(NF / 4);
    const int initBlocks = (nquads + 255) / 256;
    const int edgeBlocks = ((E * 16) + 255) / 256;
    const int strips     = (N + 15) / 16;
    const int gemmBlocks = (strips + 7) / 8;

    // k = 0 : out = x @ W0 + bias
    cheb_gemm_wmma<<<gemmBlocks, 256, 0, stream>>>(x, W, bias, out, 1, N);

    // k = 1 : T1 = L x = (-1)*x + scatter(-w_e * x[src])
    cheb_init_combine<<<initBlocks, 256, 0, stream>>>(b0, x, -1.0f, x, 0.0f, nquads);
    cheb_edge_scatter<<<edgeBlocks, 256, 0, stream>>>(b0, x, src, dst, ew, -1.0f, E);
    cheb_gemm_wmma<<<gemmBlocks, 256, 0, stream>>>(b0, W + 1 * FDIM * FDIM, bias, out, 0, N);

    // k = 2..5 : Tk = 2 L T_{k-1} - T_{k-2}
    //          = (-2)*T_{k-1} + (-1)*T_{k-2} + scatter(-2 w_e * T_{k-1}[src])
    const float* prev2 = x;
    const float* prev1 = b0;
    int curi = 1;
    for (int k = 2; k < 6; ++k) {
        float* cur = bufs[curi];
        cheb_init_combine<<<initBlocks, 256, 0, stream>>>(cur, prev1, -2.0f, prev2, -1.0f, nquads);
        cheb_edge_scatter<<<edgeBlocks, 256, 0, stream>>>(cur, prev1, src, dst, ew, -2.0f, E);
        cheb_gemm_wmma<<<gemmBlocks, 256, 0, stream>>>(cur, W + k * FDIM * FDIM, bias, out, 0, N);
        prev2 = prev1;
        prev1 = cur;
        curi  = (curi + 1) % 3;
    }
}